// GlobalEquivariantSetupAbstraction_68229850464410
// MI455X (gfx1250) — compile-verified
//
#include <hip/hip_runtime.h>
#include <hip/hip_bf16.h>

typedef __attribute__((ext_vector_type(2))) float v2f;
typedef __attribute__((ext_vector_type(4))) float v4f;
typedef __attribute__((ext_vector_type(8))) float v8f;

#define NB 2048          // number of graphs
#define C_OUT 32
#define C_IN 8
#define KDIM 64          // C_IN * 8 basis fns
#define WIDTH_K 4.0f
#define SIGMA_K 4.0f     // WIDTH / (N_RINGS-1)
#define EPS_K 1e-6f

// ---------------------------------------------------------------------------
// Kernel 1: segmented sum of coords + counts (wave-aggregated atomics).
// batch is sorted, so a wave is almost always uniform -> 1 atomic set / wave.
// ---------------------------------------------------------------------------
__global__ void pool_accum_kernel(const float* __restrict__ coords,
                                  const int* __restrict__ batch,
                                  float* __restrict__ posSum,   // [NB*3]
                                  float* __restrict__ cnt,      // [NB]
                                  int n) {
    const int lane = threadIdx.x & 31;
    const int gwave = (blockIdx.x * blockDim.x + threadIdx.x) >> 5;
    const int nW = (gridDim.x * blockDim.x) >> 5;
    for (int e0 = gwave * 32; e0 < n; e0 += nW * 32) {
        int i = e0 + lane;
        bool ok = i < n;
        int bRaw = ok ? batch[i] : 0x7fffffff;
        int bl0 = __shfl(bRaw, 0, 32);
        int b = ok ? bRaw : bl0;
        float cx = ok ? coords[i * 3 + 0] : 0.0f;
        float cy = ok ? coords[i * 3 + 1] : 0.0f;
        float cz = ok ? coords[i * 3 + 2] : 0.0f;
        float cc = ok ? 1.0f : 0.0f;
        unsigned long long m = __ballot(b == bl0);
        if ((m & 0xffffffffull) == 0xffffffffull) {
            for (int o = 16; o >= 1; o >>= 1) {
                cx += __shfl_down(cx, o, 32);
                cy += __shfl_down(cy, o, 32);
                cz += __shfl_down(cz, o, 32);
                cc += __shfl_down(cc, o, 32);
            }
            if (lane == 0) {
                atomicAdd(&posSum[b * 3 + 0], cx);
                atomicAdd(&posSum[b * 3 + 1], cy);
                atomicAdd(&posSum[b * 3 + 2], cz);
                atomicAdd(&cnt[b], cc);
            }
        } else if (ok) {
            atomicAdd(&posSum[b * 3 + 0], cx);
            atomicAdd(&posSum[b * 3 + 1], cy);
            atomicAdd(&posSum[b * 3 + 2], cz);
            atomicAdd(&cnt[b], 1.0f);
        }
    }
}

// ---------------------------------------------------------------------------
// Kernel 2: centroid = sum / max(cnt,1)
// ---------------------------------------------------------------------------
__global__ void pos_finalize_kernel(const float* __restrict__ posSum,
                                    const float* __restrict__ cnt,
                                    float* __restrict__ pos) {
    int b = blockIdx.x * blockDim.x + threadIdx.x;
    if (b < NB) {
        float inv = 1.0f / fmaxf(cnt[b], 1.0f);
        pos[b * 3 + 0] = posSum[b * 3 + 0] * inv;
        pos[b * 3 + 1] = posSum[b * 3 + 1] * inv;
        pos[b * 3 + 2] = posSum[b * 3 + 2] * inv;
    }
}

// ---------------------------------------------------------------------------
// Kernel 3 (hot, bandwidth-bound): G[g, i*8+b] = sum_{e in g} x[e,i]*basis[e,b].
// Each lane owns one edge per iteration and 64 running f32 accumulators
// (as 32 x v2f to encourage v_pk_fma_f32). Flush on graph change via a
// wave-private padded LDS transpose: conflict-free, 2 atomics per lane.
// ---------------------------------------------------------------------------
#define K3_WPB 4                // waves per block
#define LDS_STRIDE 33           // padded row stride (conflict-free)
__global__ void feat_accum_kernel(const float* __restrict__ x,
                                  const float* __restrict__ coords,
                                  const int* __restrict__ batch,
                                  const float* __restrict__ pos,
                                  float* __restrict__ G,        // [NB, 64]
                                  int n) {
    __shared__ float lds[K3_WPB][KDIM * LDS_STRIDE];
    const int lane = threadIdx.x & 31;
    const int wib = threadIdx.x >> 5;
    float* wls = &lds[wib][0];
    const int gwave = blockIdx.x * K3_WPB + wib;
    const int nW = gridDim.x * K3_WPB;

    int per = (((n + nW - 1) / nW) + 31) & ~31;  // strip length, multiple of 32
    const int eStart = gwave * per;
    const int eEnd = min(eStart + per, n);

    v2f acc[32];
    #pragma unroll
    for (int j = 0; j < 32; ++j) acc[j] = (v2f){0.0f, 0.0f};
    int curB = -1;

    auto flushFn = [&]() {
        if (curB >= 0) {
            #pragma unroll
            for (int j = 0; j < 32; ++j) {
                wls[(2 * j + 0) * LDS_STRIDE + lane] = acc[j].x;
                wls[(2 * j + 1) * LDS_STRIDE + lane] = acc[j].y;
            }
            __builtin_amdgcn_wave_barrier();      // DS ops are in-order per wave
            float s0 = 0.0f, s1 = 0.0f;
            #pragma unroll
            for (int j = 0; j < 32; ++j) {
                s0 += wls[lane * LDS_STRIDE + j];
                s1 += wls[(lane + 32) * LDS_STRIDE + j];
            }
            atomicAdd(&G[curB * KDIM + lane], s0);
            atomicAdd(&G[curB * KDIM + 32 + lane], s1);
            #pragma unroll
            for (int j = 0; j < 32; ++j) acc[j] = (v2f){0.0f, 0.0f};
        }
    };

    for (int e0 = eStart; e0 < eEnd; e0 += 32) {
        const int e = e0 + lane;
        const bool ok = e < eEnd;
        int bRaw = ok ? batch[e] : 0x7fffffff;
        int bl0 = __shfl(bRaw, 0, 32);          // lane 0 is always in range
        int b = ok ? bRaw : bl0;

        v4f xlo = (v4f){0, 0, 0, 0}, xhi = (v4f){0, 0, 0, 0};
        v2f bas2[4];
        #pragma unroll
        for (int j = 0; j < 4; ++j) bas2[j] = (v2f){0.0f, 0.0f};
        if (ok) {
            float rx = pos[b * 3 + 0] - coords[e * 3 + 0];
            float ry = pos[b * 3 + 1] - coords[e * 3 + 1];
            float rz = pos[b * 3 + 2] - coords[e * 3 + 2];
            float d = sqrtf(rx * rx + ry * ry + rz * rz);
            float t0r = d * (1.0f / SIGMA_K);
            float t1r = (d - WIDTH_K) * (1.0f / SIGMA_K);
            float r0 = __expf(-0.5f * t0r * t0r);
            float r1 = __expf(-0.5f * t1r * t1r);
            float inv = 1.0f / fmaxf(d, EPS_K);
            float dx = rx * inv, dy = ry * inv, dz = rz * inv;
            bas2[0] = (v2f){r0, r0 * dx};
            bas2[1] = (v2f){r0 * dy, r0 * dz};
            bas2[2] = (v2f){r1, r1 * dx};
            bas2[3] = (v2f){r1 * dy, r1 * dz};
            const v4f* xp = (const v4f*)(x + (size_t)e * C_IN);
            xlo = xp[0];
            xhi = xp[1];
        }
        float xv[C_IN] = {xlo.x, xlo.y, xlo.z, xlo.w, xhi.x, xhi.y, xhi.z, xhi.w};

        unsigned long long mbal = __ballot(b == bl0);
        bool uni = ((mbal & 0xffffffffull) == 0xffffffffull);
        if (uni) {
            if (bl0 != curB) { flushFn(); curB = bl0; }
            #pragma unroll
            for (int i = 0; i < C_IN; ++i)
                #pragma unroll
                for (int j = 0; j < 4; ++j)
                    acc[i * 4 + j] += bas2[j] * xv[i];   // -> v_pk_fma_f32
        } else {
            // graph boundary inside this 32-edge window: process per distinct
            // graph id in ascending order (wave-uniform sub-rounds).
            bool done = false;
            while (true) {
                int sel = done ? 0x7fffffff : b;
                int mn = sel;
                for (int o = 16; o >= 1; o >>= 1)
                    mn = min(mn, __shfl_xor(mn, o, 32));
                if (mn == 0x7fffffff) break;
                if (curB != mn) { flushFn(); curB = mn; }
                if (!done && b == mn) {
                    #pragma unroll
                    for (int i = 0; i < C_IN; ++i)
                        #pragma unroll
                        for (int j = 0; j < 4; ++j)
                            acc[i * 4 + j] += bas2[j] * xv[i];
                    done = true;
                }
            }
        }
    }
    flushFn();
}

// ---------------------------------------------------------------------------
// Kernel 4: exact-f32 WMMA GEMM  msg[2048,32] = G[2048,64] x Wmat[64,32],
// Wmat[k][o] = W[o*64 + k].  One wave per 16-row tile, K stepped by 4 with
// v_wmma_f32_16x16x4_f32; 2 N-tiles of 16 columns.
// ---------------------------------------------------------------------------
__global__ void gemm_wmma_kernel(const float* __restrict__ G,    // [NB,64]
                                 const float* __restrict__ W,    // [32,8,8]
                                 float* __restrict__ msgOut) {   // [NB,32]
    const int lane = threadIdx.x & 31;
    const bool hi = lane >= 16;
    const int lm = lane & 15;
    const int wave = (blockIdx.x * blockDim.x + threadIdx.x) >> 5;
    const int row0 = wave * 16;

    v8f c0 = {};
    v8f c1 = {};
    #pragma unroll
    for (int k0 = 0; k0 < KDIM; k0 += 4) {
        // A 16x4 f32 frag: lanes<16 hold K=k0,k0+1; lanes>=16 hold K=k0+2,k0+3
        int ka = k0 + (hi ? 2 : 0);
        v2f a;
        a.x = G[(row0 + lm) * KDIM + ka + 0];
        a.y = G[(row0 + lm) * KDIM + ka + 1];
        // B 4x16 f32 frag: VGPR0 rows K=k0 (lanes<16) / k0+1 (lanes>=16),
        //                  VGPR1 rows K=k0+2 / k0+3;  N = lm
        int kb0 = k0 + (hi ? 1 : 0);
        int kb1 = k0 + 2 + (hi ? 1 : 0);
        v2f b0, b1;
        b0.x = W[lm * KDIM + kb0];
        b0.y = W[lm * KDIM + kb1];
        b1.x = W[(16 + lm) * KDIM + kb0];
        b1.y = W[(16 + lm) * KDIM + kb1];
        c0 = __builtin_amdgcn_wmma_f32_16x16x4_f32(false, a, false, b0,
                                                   (short)0, c0, false, false);
        c1 = __builtin_amdgcn_wmma_f32_16x16x4_f32(false, a, false, b1,
                                                   (short)0, c1, false, false);
    }
    // D layout: VGPR v -> row (v + 8*hi), column lm
    #pragma unroll
    for (int v = 0; v < 8; ++v) {
        int row = row0 + v + (hi ? 8 : 0);
        msgOut[row * C_OUT + lm] = c0[v];
        msgOut[row * C_OUT + 16 + lm] = c1[v];
    }
}

// ---------------------------------------------------------------------------
// Kernel 5: per-channel batch-norm (population stats over B=2048) + NormReLU.
// ---------------------------------------------------------------------------
__global__ void bn_act_kernel(const float* __restrict__ msg,
                              const float* __restrict__ gamma,
                              const float* __restrict__ beta,
                              const float* __restrict__ act_bias,
                              float* __restrict__ y) {
    const int c = blockIdx.x;       // channel
    const int tid = threadIdx.x;    // 256 threads
    float vals[8];
    float s = 0.0f, sq = 0.0f;
    #pragma unroll
    for (int j = 0; j < 8; ++j) {
        float v = msg[(tid + j * 256) * C_OUT + c];
        vals[j] = v;
        s += v;
        sq += v * v;
    }
    __shared__ float shs[256];
    __shared__ float shq[256];
    shs[tid] = s; shq[tid] = sq;
    __syncthreads();
    for (int o = 128; o > 0; o >>= 1) {
        if (tid < o) { shs[tid] += shs[tid + o]; shq[tid] += shq[tid + o]; }
        __syncthreads();
    }
    float mean = shs[0] * (1.0f / (float)NB);
    float var = shq[0] * (1.0f / (float)NB) - mean * mean;
    float istd = 1.0f / sqrtf(var + 1e-5f);
    float g = gamma[c], bt = beta[c], ab = act_bias[c];
    #pragma unroll
    for (int j = 0; j < 8; ++j) {
        float v = (vals[j] - mean) * istd * g + bt;
        y[(tid + j * 256) * C_OUT + c] = fmaxf(fabsf(v) - ab, 0.0f);
    }
}

// ---------------------------------------------------------------------------
extern "C" void kernel_launch(void* const* d_in, const int* in_sizes, int n_in,
                              void* d_out, int out_size, void* d_ws, size_t ws_size,
                              hipStream_t stream) {
    const float* x        = (const float*)d_in[0];
    const float* coords   = (const float*)d_in[1];
    const int*   batch    = (const int*)d_in[2];
    const float* W        = (const float*)d_in[3];
    const float* gamma    = (const float*)d_in[4];
    const float* beta     = (const float*)d_in[5];
    const float* act_bias = (const float*)d_in[6];
    float* out = (float*)d_out;

    const int n = in_sizes[1] / 3;   // N points

    // workspace (floats): posSum[3B] | cnt[B] | pos[3B] | G[B*64] | msg[B*32]
    float* ws     = (float*)d_ws;
    float* posSum = ws;
    float* cnt    = ws + 3 * NB;
    float* pos    = ws + 4 * NB;
    float* G      = ws + 7 * NB;
    float* msg    = ws + 7 * NB + NB * KDIM;

    // zero posSum, cnt, pos, G (msg is fully overwritten)
    hipMemsetAsync(d_ws, 0, (size_t)(7 * NB + NB * KDIM) * sizeof(float), stream);

    pool_accum_kernel<<<1024, 256, 0, stream>>>(coords, batch, posSum, cnt, n);
    pos_finalize_kernel<<<NB / 256, 256, 0, stream>>>(posSum, cnt, pos);
    feat_accum_kernel<<<1024, 32 * K3_WPB, 0, stream>>>(x, coords, batch, pos, G, n);
    gemm_wmma_kernel<<<(NB / 16) / 8, 256, 0, stream>>>(G, W, msg);
    bn_act_kernel<<<C_OUT, 256, 0, stream>>>(msg, gamma, beta, act_bias, out);
}